// GATConvLayer_78847009620725
// MI455X (gfx1250) — compile-verified
//
#include <hip/hip_runtime.h>
#include <hip/hip_bf16.h>

// ---------------------------------------------------------------------------
// GAT layer for MI455X (gfx1250, wave32).
//   N=50000 nodes, E=800000 edges, F_IN=128, H=4 heads, D=32 (H*D=128)
// Pipeline:
//   1) h = x @ W            (fp32 WMMA 16x16x4, bandwidth-bound)
//   2) alpha_l/alpha_r      (per node,head 32-dots)
//   3) global per-head max  (ordered-uint atomicMax)
//   4) softmax denominators (f32 atomics into denom[N,H])
//   5) weighted scatter-add (wave per edge, 128 f32 atomics into L2-resident out)
// ---------------------------------------------------------------------------

typedef __attribute__((ext_vector_type(2))) float v2f;
typedef __attribute__((ext_vector_type(8))) float v8f;

// Monotonic float <-> uint mapping so atomicMax(uint) == float max.
__device__ __forceinline__ unsigned f2ord(float f) {
    unsigned u = __float_as_uint(f);
    return (u & 0x80000000u) ? ~u : (u | 0x80000000u);
}
__device__ __forceinline__ float ord2f(unsigned u) {
    unsigned v = (u & 0x80000000u) ? (u & 0x7FFFFFFFu) : ~u;
    return __uint_as_float(v);
}

// ---------------------------------------------------------------------------
// 1) h = x @ W   x:[N,128] row-major, W:[128,128] row-major, h:[N,128]
// Block = 256 thr = 8 waves. Block b -> rows [16b,16b+16), wave w -> cols
// [16w,16w+16). K-loop of 32 x V_WMMA_F32_16X16X4_F32.
// A 16x4 layout (ISA 7.12.2): lanes 0-15: M=lane, V0=K0,V1=K1;
//                             lanes 16-31: M=lane-16, V0=K2,V1=K3.
// B 4x16 mirrors A with N on lanes. C/D: VGPR r -> M=r (lanes0-15) / r+8.
// N must be a multiple of 16 (50000 = 3125*16). EXEC all-ones: no divergence.
// ---------------------------------------------------------------------------
__global__ void gat_gemm_wmma(const float* __restrict__ x,
                              const float* __restrict__ W,
                              float* __restrict__ h) {
    const int wave   = threadIdx.x >> 5;     // 0..7 -> 16-col tile
    const int lane   = threadIdx.x & 31;
    const int lhalf  = lane & 15;
    const int khalf  = (lane >> 4) << 1;     // 0 or 2
    const int rowA   = blockIdx.x * 16 + lhalf;
    const int colB   = wave * 16 + lhalf;

    const float* __restrict__ arow = x + (size_t)rowA * 128;

    v8f c = {0.f, 0.f, 0.f, 0.f, 0.f, 0.f, 0.f, 0.f};

#pragma unroll 4
    for (int k0 = 0; k0 < 128; k0 += 4) {
        const int k = k0 + khalf;
        v2f a;
        a.x = arow[k];
        a.y = arow[k + 1];
        v2f b;
        b.x = W[(size_t)k * 128 + colB];
        b.y = W[(size_t)(k + 1) * 128 + colB];
        // (neg_a, A, neg_b, B, c_mod, C, reuse_a, reuse_b)
        c = __builtin_amdgcn_wmma_f32_16x16x4_f32(false, a, false, b,
                                                  (short)0, c, false, false);
    }

    const int mhi = (lane >> 4) << 3;        // 0 or 8
#pragma unroll
    for (int r = 0; r < 8; ++r) {
        const int m = blockIdx.x * 16 + r + mhi;
        h[(size_t)m * 128 + wave * 16 + lhalf] = c[r];
    }
}

// ---------------------------------------------------------------------------
// init: out[n,j] = bias[j]; denom = 0; maxbuf = encode(-3e38)
// ---------------------------------------------------------------------------
__global__ void gat_init(float* __restrict__ out, const float* __restrict__ bias,
                         float* __restrict__ denom, unsigned* __restrict__ maxbuf,
                         int total, int nh) {
    const int i = blockIdx.x * blockDim.x + threadIdx.x;
    if (i < 4)     maxbuf[i] = f2ord(-3.0e38f);
    if (i < nh)    denom[i]  = 0.0f;
    if (i < total) out[i]    = bias[i & 127];
}

// ---------------------------------------------------------------------------
// 2) alpha_l[n,h] = sum_d h[n,h,d]*att_l[h,d]  (same for att_r)
// ---------------------------------------------------------------------------
__global__ void gat_alpha(const float* __restrict__ h,
                          const float* __restrict__ att_l,
                          const float* __restrict__ att_r,
                          float* __restrict__ alpha_l,
                          float* __restrict__ alpha_r, int nh) {
    const int i = blockIdx.x * blockDim.x + threadIdx.x;   // i = n*4 + head
    if (i >= nh) return;
    const int head = i & 3;
    const float* __restrict__ hv = h + ((size_t)(i >> 2)) * 128 + head * 32;
    const float* __restrict__ al = att_l + head * 32;
    const float* __restrict__ ar = att_r + head * 32;
    float sl = 0.f, sr = 0.f;
#pragma unroll
    for (int d = 0; d < 32; ++d) {
        const float v = hv[d];
        sl += v * al[d];
        sr += v * ar[d];
    }
    alpha_l[i] = sl;
    alpha_r[i] = sr;
}

// ---------------------------------------------------------------------------
// 3) global per-head max of leaky_relu(alpha_l[from]+alpha_r[to]) over edges
// ---------------------------------------------------------------------------
__global__ void gat_edge_max(const int* __restrict__ ei,
                             const float* __restrict__ alpha_l,
                             const float* __restrict__ alpha_r,
                             unsigned* __restrict__ maxbuf, int E) {
    __shared__ unsigned smax[4];
    if (threadIdx.x < 4) smax[threadIdx.x] = 0u;   // encodes below any real value
    __syncthreads();
    const int e = blockIdx.x * blockDim.x + threadIdx.x;
    if (e < E) {
        const int f = ei[e];
        const int t = ei[E + e];
#pragma unroll
        for (int hd = 0; hd < 4; ++hd) {
            float a = alpha_l[f * 4 + hd] + alpha_r[t * 4 + hd];
            a = a > 0.f ? a : 0.2f * a;
            atomicMax(&smax[hd], f2ord(a));
        }
    }
    __syncthreads();
    if (threadIdx.x < 4) atomicMax(&maxbuf[threadIdx.x], smax[threadIdx.x]);
}

// ---------------------------------------------------------------------------
// 4) denom[to,h] += exp(att - max[h])
// ---------------------------------------------------------------------------
__global__ void gat_denom(const int* __restrict__ ei,
                          const float* __restrict__ alpha_l,
                          const float* __restrict__ alpha_r,
                          const unsigned* __restrict__ maxbuf,
                          float* __restrict__ denom, int E) {
    const int e = blockIdx.x * blockDim.x + threadIdx.x;
    if (e >= E) return;
    const int f = ei[e];
    const int t = ei[E + e];
#pragma unroll
    for (int hd = 0; hd < 4; ++hd) {
        float a = alpha_l[f * 4 + hd] + alpha_r[t * 4 + hd];
        a = a > 0.f ? a : 0.2f * a;
        const float ex = __expf(a - ord2f(maxbuf[hd]));
        atomicAdd(&denom[t * 4 + hd], ex);
    }
}

// ---------------------------------------------------------------------------
// 5) out[to, h*32+d] += w[e,h] * hbuf[from, h*32+d]   (one wave per edge;
//    lane = d, loop over 4 heads). h and out are L2-resident (25.6 MB each).
// ---------------------------------------------------------------------------
__global__ void gat_aggregate(const float* __restrict__ hbuf,
                              const int* __restrict__ ei,
                              const float* __restrict__ alpha_l,
                              const float* __restrict__ alpha_r,
                              const float* __restrict__ denom,
                              const unsigned* __restrict__ maxbuf,
                              float* __restrict__ out, int E) {
    const int gw   = (int)((blockIdx.x * (unsigned)blockDim.x + threadIdx.x) >> 5);
    const int lane = threadIdx.x & 31;
    if (gw >= E) return;
    const int f = ei[gw];
    const int t = ei[E + gw];
    const float* __restrict__ src = hbuf + (size_t)f * 128;
    float* __restrict__ dst = out + (size_t)t * 128;
#pragma unroll
    for (int hd = 0; hd < 4; ++hd) {
        float a = alpha_l[f * 4 + hd] + alpha_r[t * 4 + hd];      // broadcast loads
        a = a > 0.f ? a : 0.2f * a;
        const float ex = __expf(a - ord2f(maxbuf[hd]));
        const float w  = ex / (denom[t * 4 + hd] + 1e-9f);
        atomicAdd(&dst[hd * 32 + lane], w * src[hd * 32 + lane]);
    }
}

// ---------------------------------------------------------------------------
// Launch
// ---------------------------------------------------------------------------
extern "C" void kernel_launch(void* const* d_in, const int* in_sizes, int n_in,
                              void* d_out, int out_size, void* d_ws, size_t ws_size,
                              hipStream_t stream) {
    const float* x     = (const float*)d_in[0];     // [N,128]
    const int*   ei    = (const int*)d_in[1];       // [2,E]
    const float* W     = (const float*)d_in[2];     // [128,128]
    const float* att_l = (const float*)d_in[3];     // [4,32]
    const float* att_r = (const float*)d_in[4];     // [4,32]
    const float* bias  = (const float*)d_in[5];     // [128]
    float* out = (float*)d_out;                     // [N,128]

    const int N  = in_sizes[0] / 128;               // 50000
    const int E  = in_sizes[1] / 2;                 // 800000
    const int NH = N * 4;

    // Workspace layout
    float* h       = (float*)d_ws;                  // N*128 floats (25.6 MB)
    float* alpha_l = h + (size_t)N * 128;           // N*4
    float* alpha_r = alpha_l + NH;                  // N*4
    float* denom   = alpha_r + NH;                  // N*4
    unsigned* maxbuf = (unsigned*)(denom + NH);     // 4

    const int BLK = 256;

    // init out=bias, denom=0, maxbuf=-inf
    {
        const int total = N * 128;
        gat_init<<<(total + BLK - 1) / BLK, BLK, 0, stream>>>(out, bias, denom,
                                                              maxbuf, total, NH);
    }
    // h = x @ W  (N/16 blocks of 8 waves; N % 16 == 0)
    gat_gemm_wmma<<<N / 16, BLK, 0, stream>>>(x, W, h);
    // per-node attention logits
    gat_alpha<<<(NH + BLK - 1) / BLK, BLK, 0, stream>>>(h, att_l, att_r,
                                                        alpha_l, alpha_r, NH);
    // global per-head max over edges
    gat_edge_max<<<(E + BLK - 1) / BLK, BLK, 0, stream>>>(ei, alpha_l, alpha_r,
                                                          maxbuf, E);
    // softmax denominators
    gat_denom<<<(E + BLK - 1) / BLK, BLK, 0, stream>>>(ei, alpha_l, alpha_r,
                                                       maxbuf, denom, E);
    // weighted scatter aggregation: one wave per edge
    {
        const long long threads = (long long)E * 32;
        const int blocks = (int)((threads + BLK - 1) / BLK);
        gat_aggregate<<<blocks, BLK, 0, stream>>>(h, ei, alpha_l, alpha_r,
                                                  denom, maxbuf, out, E);
    }
}